// Node2Vec_12824772346469
// MI455X (gfx1250) — compile-verified
//
#include <hip/hip_runtime.h>
#include <hip/hip_bf16.h>
#include <math.h>

typedef __attribute__((ext_vector_type(2))) float v2f;
typedef __attribute__((ext_vector_type(8))) float v8f;

#define DIM      128
#define LWALK    5
#define WIN      (LWALK + 1)   // 6
#define ROWLEN   16
#define BATCH    4096
#define WAVES_PER_BLOCK 8

// One wave32 per batch row. Loads the 16x128 embedding tile directly into the
// V_WMMA_F32_16X16X4_F32 A-operand layout and computes the 16x16 Gram matrix
// G = emb * emb^T via 32 accumulating WMMAs (A and B share registers since the
// striped layouts coincide for the transpose). dots[j] = G[j][0].
__global__ __launch_bounds__(256) void node2vec_loss_kernel(
    const float* __restrict__ X,
    const int*  __restrict__ rw,
    float*      __restrict__ per_row)
{
    const int lane = threadIdx.x & 31;
    const int wave = threadIdx.x >> 5;
    const int b    = blockIdx.x * WAVES_PER_BLOCK + wave;

    // WMMA A-operand striping: row m = lane&15; lanes 0-15 carry K pair {k0,k0+1},
    // lanes 16-31 carry {k0+2,k0+3}.
    const int m    = lane & 15;
    const int ksub = (lane >> 4) << 1;

    const long long node = (long long)rw[b * ROWLEN + m];
    const float* rowp = X + node * (long long)DIM + ksub;

    // Gather the whole tile: 32 x float2 per lane (16B contiguous per row per
    // lane pair) -> global_load_b64, fully coalesced over the 16 rows.
    v2f a[32];
    #pragma unroll
    for (int i = 0; i < 32; ++i) {
        a[i] = *(const v2f*)(rowp + 4 * i);
    }

    // 32 accumulating f32 WMMAs: D = A * A^T + C over K = 128.
    v8f c = {};
    #pragma unroll
    for (int i = 0; i < 32; ++i) {
        c = __builtin_amdgcn_wmma_f32_16x16x4_f32(
            /*neg_a=*/false, a[i],
            /*neg_b=*/false, a[i],
            /*c_mod=*/(short)0, c,
            /*reuse_a=*/false, /*reuse_b=*/false);
    }

    // Column N=0 of D: lane 0 holds G[0..7][0] in c[0..7], lane 16 holds G[8..15][0].
    float dots[16];
    #pragma unroll
    for (int v = 0; v < 8; ++v) {
        dots[v]     = __shfl(c[v], 0, 32);
        dots[v + 8] = __shfl(c[v], 16, 32);
    }

    // numerator = sum dots[1..5]
    float numer = 0.0f;
    #pragma unroll
    for (int j = 1; j < WIN; ++j) numer += dots[j];

    // duplicate mask over the first 6 walk indices
    int w[WIN];
    #pragma unroll
    for (int j = 0; j < WIN; ++j) w[j] = rw[b * ROWLEN + j];

    float denom = 0.0f;
    #pragma unroll
    for (int j = 0; j < WIN; ++j) {
        bool dup = false;
        #pragma unroll
        for (int k = 0; k < WIN; ++k) {
            if (k < j) dup = dup || (w[k] == w[j]);
        }
        if (!dup) denom += expf(dots[j]);
    }
    #pragma unroll
    for (int j = WIN; j < ROWLEN; ++j) denom += expf(dots[j]);

    const float loss = (float)LWALK * logf(denom) - numer;

    if (lane == 0) per_row[b] = loss;
}

// Deterministic fixed-order reduction: 4096 partial losses -> mean.
__global__ __launch_bounds__(256) void reduce_mean_kernel(
    const float* __restrict__ per_row,
    float*       __restrict__ out)
{
    __shared__ float s[256];
    const int t = threadIdx.x;
    float acc = 0.0f;
    for (int i = t; i < BATCH; i += 256) acc += per_row[i];
    s[t] = acc;
    __syncthreads();
    for (int off = 128; off > 0; off >>= 1) {
        if (t < off) s[t] += s[t + off];
        __syncthreads();
    }
    if (t == 0) out[0] = s[0] * (1.0f / (float)BATCH);
}

extern "C" void kernel_launch(void* const* d_in, const int* in_sizes, int n_in,
                              void* d_out, int out_size, void* d_ws, size_t ws_size,
                              hipStream_t stream) {
    const float* X  = (const float*)d_in[0];   // (500000, 128) f32
    const int*   rw = (const int*)d_in[1];     // (4096, 16) walk indices
    // d_in[2] is the scalar l (== 5), baked in as LWALK.

    float* out     = (float*)d_out;            // scalar f32 mean loss
    float* per_row = (float*)d_ws;             // BATCH floats of scratch

    node2vec_loss_kernel<<<BATCH / WAVES_PER_BLOCK, 256, 0, stream>>>(X, rw, per_row);
    reduce_mean_kernel<<<1, 256, 0, stream>>>(per_row, out);
}